// ZBLPotential_79860621902322
// MI455X (gfx1250) — compile-verified
//
#include <hip/hip_runtime.h>

#define N_ATOMS      1000000
#define N_EDGES      16000000
#define NUM_SYSTEMS  10000
#define MAX_Z        97

// Constants (compile-time folded)
constexpr float COULOMB = 138.9354576f;
constexpr float A_PREF  = (float)(0.8854 * 0.0529177210903);   // nm
constexpr float RCP_A   = 1.0f / A_PREF;
constexpr float LOG2E   = 1.4426950408889634f;
constexpr float C1 = -3.2f    * LOG2E;
constexpr float C2 = -0.9423f * LOG2E;
constexpr float C3 = -0.4029f * LOG2E;
constexpr float C4 = -0.2016f * LOG2E;

// ---------------------------------------------------------------------------
// Pack per-atom data: one int32 = (subsys << 7) | z  (z in [1,96] fits 7 bits,
// subsys < 10000 fits 14 bits). Cuts per-edge gathers from 3 to 2.
// ---------------------------------------------------------------------------
__global__ __launch_bounds__(256) void zbl_pack_kernel(
    const int* __restrict__ zn, const int* __restrict__ subsys,
    int* __restrict__ packed)
{
    const int a = blockIdx.x * blockDim.x + threadIdx.x;
    if (a < N_ATOMS) packed[a] = (subsys[a] << 7) | zn[a];
}

// ---------------------------------------------------------------------------
// Main kernel: grid-stride over edges, LDS-binned segment accumulation.
// LDS: 10000 f32 accumulator (40 KB) + 97 x float2 {z^0.23, radius} table.
// ~40.8 KB LDS -> 8 blocks resident per 320KB WGP (64 waves/WGP) so the
// random L2 gathers are latency-hidden by occupancy, not by async-LDS.
// ---------------------------------------------------------------------------
template <bool PACKED>
__global__ __launch_bounds__(256) void zbl_main_kernel(
    const float* __restrict__ d_ij,      // [N_EDGES]
    const float* __restrict__ radii,     // [MAX_Z]
    const int*   __restrict__ pairs,     // [2, N_EDGES] row-major
    const int*   __restrict__ zn,        // [N_ATOMS]        (!PACKED)
    const int*   __restrict__ subsys,    // [N_ATOMS]        (!PACKED)
    const int*   __restrict__ packed,    // [N_ATOMS]        (PACKED)
    float*       __restrict__ partials,  // [G, NUM_SYSTEMS] (partial path)
    float*       __restrict__ out_at,    // [NUM_SYSTEMS]    (atomic path)
    int use_atomic)
{
    __shared__ float  s_acc[NUM_SYSTEMS];
    __shared__ float2 s_tab[MAX_Z];      // {z^0.23, radius}

    const int tid = threadIdx.x;

    // Per-block tables: z^0.23 = exp2(0.23 * log2(z)) via v_log/v_exp.
    if (tid < MAX_Z) {
        float zf = (float)tid;
        float zp = (tid == 0) ? 0.0f
                 : __builtin_amdgcn_exp2f(0.23f * __builtin_amdgcn_logf(zf));
        s_tab[tid] = make_float2(zp, radii[tid]);
    }
    for (int s = tid; s < NUM_SYSTEMS; s += blockDim.x) s_acc[s] = 0.0f;
    __syncthreads();

    // Streaming pointers, 4 edges per thread per iteration (B128 loads).
    const int4*   pi4 = (const int4*)(pairs);            // idx_i row
    const int4*   pj4 = (const int4*)(pairs + N_EDGES);  // idx_j row
    const float4* dd4 = (const float4*)(d_ij);

    const int nvec    = N_EDGES >> 2;                    // 4M int4 chunks
    const int vbase   = blockIdx.x * blockDim.x + tid;
    const int vstride = gridDim.x * blockDim.x;

    for (int v = vbase; v < nvec; v += vstride) {
        int4   ii = pi4[v];
        int4   jj = pj4[v];
        float4 dd = dd4[v];

        // gfx1250 global_prefetch_b8: pull next grid-stride chunk.
        {
            int vp = v + vstride;   // speculative; OOB prefetch is dropped
            __builtin_prefetch(&pi4[vp], 0, 0);
            __builtin_prefetch(&pj4[vp], 0, 0);
            __builtin_prefetch(&dd4[vp], 0, 0);
        }

        const int   ia[4] = {ii.x, ii.y, ii.z, ii.w};
        const int   ja[4] = {jj.x, jj.y, jj.z, jj.w};
        const float da[4] = {dd.x, dd.y, dd.z, dd.w};

#pragma unroll
        for (int k = 0; k < 4; ++k) {
            const int   i = ia[k];
            const int   j = ja[k];
            const float d = da[k];
            if (i < j) {                               // edge mask
                int pi = 0, zi, zj;
                if (PACKED) {
                    pi = packed[i];                    // z + seg in one gather
                    const int pj = packed[j];
                    zi = pi & 127;
                    zj = pj & 127;
                } else {
                    zi = zn[i];
                    zj = zn[j];
                }
                const float2 ti = s_tab[zi];           // one ds_load_b64 each
                const float2 tj = s_tab[zj];
                const float rsum = ti.y + tj.y;
                if (d < rsum) {                        // phi == 0 otherwise
                    const int   seg = PACKED ? (pi >> 7) : subsys[i];
                    const float x   = d * (ti.x + tj.x) * RCP_A;
                    const float f   = 0.1818f  * __builtin_amdgcn_exp2f(C1 * x)
                                    + 0.5099f  * __builtin_amdgcn_exp2f(C2 * x)
                                    + 0.2802f  * __builtin_amdgcn_exp2f(C3 * x)
                                    + 0.02817f * __builtin_amdgcn_exp2f(C4 * x);
                    // cos(pi*d/rsum): v_cos_f32 computes cos(2*pi*arg);
                    // arg = 0.5*d/rsum in [0, 0.5) -> no range reduction.
                    const float arg = 0.5f * d * __builtin_amdgcn_rcpf(rsum);
                    const float phi = 0.5f * (__builtin_amdgcn_cosf(arg) + 1.0f);
                    const float zz  = (float)(zi * zj);
                    const float e   = f * phi * COULOMB * zz
                                    * __builtin_amdgcn_rcpf(d);
                    atomicAdd(&s_acc[seg], e);         // ds_add_f32
                }
            }
        }
    }
    __syncthreads();

    if (use_atomic) {
        for (int s = tid; s < NUM_SYSTEMS; s += blockDim.x) {
            const float vsum = s_acc[s];
            if (vsum != 0.0f) atomicAdd(&out_at[s], vsum);
        }
    } else {
        float* dst = partials + (size_t)blockIdx.x * NUM_SYSTEMS;
        for (int s = tid; s < NUM_SYSTEMS; s += blockDim.x) dst[s] = s_acc[s];
    }
}

// ---------------------------------------------------------------------------
// Reduce G partial vectors -> out. float4 per thread; for each g the loads of
// adjacent threads are contiguous (fully coalesced). ~G*40KB total traffic.
// ---------------------------------------------------------------------------
__global__ __launch_bounds__(256) void zbl_reduce_kernel(
    const float4* __restrict__ partials4, float4* __restrict__ out4, int G)
{
    const int s4 = blockIdx.x * blockDim.x + threadIdx.x;  // NUM_SYSTEMS/4
    if (s4 < NUM_SYSTEMS / 4) {
        float4 acc = make_float4(0.f, 0.f, 0.f, 0.f);
        for (int g = 0; g < G; ++g) {
            const float4 p = partials4[(size_t)g * (NUM_SYSTEMS / 4) + s4];
            acc.x += p.x; acc.y += p.y; acc.z += p.z; acc.w += p.w;
        }
        out4[s4] = acc;
    }
}

__global__ __launch_bounds__(256) void zbl_zero_kernel(float* __restrict__ out)
{
    const int s = blockIdx.x * blockDim.x + threadIdx.x;
    if (s < NUM_SYSTEMS) out[s] = 0.0f;
}

// ---------------------------------------------------------------------------
extern "C" void kernel_launch(void* const* d_in, const int* in_sizes, int n_in,
                              void* d_out, int out_size, void* d_ws, size_t ws_size,
                              hipStream_t stream)
{
    const float* d_ij   = (const float*)d_in[0];   // local_d_ij          f32[16M]
    const float* radii  = (const float*)d_in[1];   // radii_table         f32[97]
    const int*   pairs  = (const int*)  d_in[2];   // local_pair_indices  i32[2,16M]
    const int*   zn     = (const int*)  d_in[3];   // atomic_numbers      i32[1M]
    const int*   subsys = (const int*)  d_in[4];   // atomic_subsystem_.. i32[1M]
    float*       out    = (float*)d_out;           // f32[10000]

    const size_t PACK_BYTES    = (size_t)N_ATOMS * sizeof(int);
    const size_t PARTIAL_BYTES = (size_t)NUM_SYSTEMS * sizeof(float);

    const bool can_pack = ws_size >= PACK_BYTES;
    int* packed = can_pack ? (int*)d_ws : nullptr;
    const size_t rem = can_pack ? (ws_size - PACK_BYTES) : ws_size;
    float* partials = can_pack ? (float*)((char*)d_ws + PACK_BYTES)
                               : (float*)d_ws;

    int Gmax = (int)(rem / PARTIAL_BYTES);
    int G    = Gmax < 768 ? Gmax : 768;
    const bool use_partials = G >= 16;

    if (can_pack)
        zbl_pack_kernel<<<(N_ATOMS + 255) / 256, 256, 0, stream>>>(
            zn, subsys, packed);

    if (use_partials) {
        if (can_pack)
            zbl_main_kernel<true><<<G, 256, 0, stream>>>(
                d_ij, radii, pairs, zn, subsys, packed, partials, nullptr, 0);
        else
            zbl_main_kernel<false><<<G, 256, 0, stream>>>(
                d_ij, radii, pairs, zn, subsys, nullptr, partials, nullptr, 0);
        zbl_reduce_kernel<<<(NUM_SYSTEMS / 4 + 255) / 256, 256, 0, stream>>>(
            (const float4*)partials, (float4*)out, G);
    } else {
        zbl_zero_kernel<<<(NUM_SYSTEMS + 255) / 256, 256, 0, stream>>>(out);
        if (can_pack)
            zbl_main_kernel<true><<<512, 256, 0, stream>>>(
                d_ij, radii, pairs, zn, subsys, packed, nullptr, out, 1);
        else
            zbl_main_kernel<false><<<512, 256, 0, stream>>>(
                d_ij, radii, pairs, zn, subsys, nullptr, nullptr, out, 1);
    }
}